// topKLoss_62517543960603
// MI455X (gfx1250) — compile-verified
//
#include <hip/hip_runtime.h>

#define KTOP 10
#define TPB  256
#define TILE 4096          // floats per LDS tile (16 KB)
#define STEPF 0.1f

typedef int v4i __attribute__((ext_vector_type(4)));
typedef __attribute__((address_space(1))) v4i v4i_glob;   // global int4
typedef __attribute__((address_space(3))) v4i v4i_lds;    // LDS int4

__device__ __forceinline__ void wait_async0() {
#if __has_builtin(__builtin_amdgcn_s_wait_asynccnt)
  __builtin_amdgcn_s_wait_asynccnt(0);
#else
  asm volatile("s_wait_asynccnt 0" ::: "memory");
#endif
}

__device__ __forceinline__ void async_copy16(const float* g, float* lds_dst) {
#if __has_builtin(__builtin_amdgcn_global_load_async_to_lds_b128)
  __builtin_amdgcn_global_load_async_to_lds_b128(
      (v4i_glob*)g, (v4i_lds*)lds_dst, 0, 0);
#else
  *(float4*)lds_dst = *(const float4*)g;   // synchronous fallback
#endif
}

// Issue one 16KB tile: 256 threads x 4 x b128 = 4096 floats, coalesced.
__device__ __forceinline__ void issue_tile(const float* rowp, int t,
                                           float* dst, int tid) {
  const float* g = rowp + (size_t)t * TILE;
#pragma unroll
  for (int i = 0; i < TILE / (TPB * 4); ++i) {
    async_copy16(g + tid * 4 + i * (TPB * 4), dst + tid * 4 + i * (TPB * 4));
  }
}

__global__ __launch_bounds__(TPB) void topk_row_kernel(
    const float* __restrict__ inp, const float* __restrict__ target,
    float* __restrict__ row_err, int C) {
  __shared__ float tile[2][TILE];        // 32 KB double buffer
  __shared__ float cv[TPB * KTOP];       // candidate values
  __shared__ int   ci[TPB * KTOP];       // candidate indices
  __shared__ float rv[TPB];              // reduction: value
  __shared__ int   ridx[TPB];            // reduction: global index
  __shared__ int   rslot[TPB];           // reduction: candidate slot

  const int row = blockIdx.x;
  const int tid = threadIdx.x;
  const float* rowp = inp + (size_t)row * (size_t)C;
  const int ntiles = C / TILE;

  // per-thread sorted (desc) top-10 in registers
  float tv[KTOP];
  int   ti[KTOP];
#pragma unroll
  for (int j = 0; j < KTOP; ++j) { tv[j] = -__builtin_inff(); ti[j] = 0; }

  // prologue: fetch tile 0
  issue_tile(rowp, 0, &tile[0][0], tid);
  wait_async0();
  __syncthreads();

  for (int t = 0; t < ntiles; ++t) {
    if (t + 1 < ntiles)
      issue_tile(rowp, t + 1, &tile[(t + 1) & 1][0], tid);  // prefetch

    const float* buf = &tile[t & 1][0];
    const int base = t * TILE;
#pragma unroll
    for (int i = 0; i < TILE / TPB; ++i) {
      const int col = i * TPB + tid;          // bank-conflict-free LDS read
      const float v = buf[col];
      if (v > tv[KTOP - 1]) {
        tv[KTOP - 1] = v;
        ti[KTOP - 1] = base + col;
        // single bubble pass (array was sorted except last) - all static idx
#pragma unroll
        for (int j = KTOP - 1; j > 0; --j) {
          if (tv[j] > tv[j - 1]) {
            float fv = tv[j]; tv[j] = tv[j - 1]; tv[j - 1] = fv;
            int   fi = ti[j]; ti[j] = ti[j - 1]; ti[j - 1] = fi;
          }
        }
      }
    }

    if (t + 1 < ntiles) wait_async0();   // prefetch landed (this wave)
    __syncthreads();                     // ...and visible to all waves
  }

  // dump per-thread candidates to LDS
#pragma unroll
  for (int j = 0; j < KTOP; ++j) {
    cv[tid * KTOP + j] = tv[j];
    ci[tid * KTOP + j] = ti[j];
  }
  __syncthreads();

  // 10 rounds of block-wide argmax with removal
  float pred = 0.0f;
  for (int k = 0; k < KTOP; ++k) {
    float bv = -__builtin_inff();
    int bi = 0, bs = tid * KTOP;
#pragma unroll
    for (int j = 0; j < KTOP; ++j) {
      const float v = cv[tid * KTOP + j];
      if (v > bv) { bv = v; bi = ci[tid * KTOP + j]; bs = tid * KTOP + j; }
    }
    rv[tid] = bv; ridx[tid] = bi; rslot[tid] = bs;
    __syncthreads();
    for (int s = TPB / 2; s > 0; s >>= 1) {
      if (tid < s) {
        if (rv[tid + s] > rv[tid]) {
          rv[tid] = rv[tid + s];
          ridx[tid] = ridx[tid + s];
          rslot[tid] = rslot[tid + s];
        }
      }
      __syncthreads();
    }
    if (tid == 0) {
      pred += rv[0] * ((float)ridx[0] * STEPF + 1.0f);
      cv[rslot[0]] = -__builtin_inff();   // remove winner
    }
    __syncthreads();
  }

  if (tid == 0) {
    const float e = target[row] - pred;
    row_err[row] = e * e;
  }
}

__global__ __launch_bounds__(256) void mean_reduce_kernel(
    const float* __restrict__ err, float* __restrict__ out, int N) {
  __shared__ float sm[256];
  const int tid = threadIdx.x;
  float s = 0.0f;
  for (int i = tid; i < N; i += 256) s += err[i];
  sm[tid] = s;
  __syncthreads();
  for (int st = 128; st > 0; st >>= 1) {
    if (tid < st) sm[tid] += sm[tid + st];
    __syncthreads();
  }
  if (tid == 0) out[0] = sm[0] / (float)N;
}

extern "C" void kernel_launch(void* const* d_in, const int* in_sizes, int n_in,
                              void* d_out, int out_size, void* d_ws, size_t ws_size,
                              hipStream_t stream) {
  const float* inp = (const float*)d_in[0];   // [N, C] fp32
  const float* tgt = (const float*)d_in[1];   // [N]    fp32
  const int N = in_sizes[1];
  const int C = in_sizes[0] / N;              // 32768, multiple of TILE
  float* row_err = (float*)d_ws;              // N floats of scratch

  topk_row_kernel<<<N, TPB, 0, stream>>>(inp, tgt, row_err, C);
  mean_reduce_kernel<<<1, 256, 0, stream>>>(row_err, (float*)d_out, N);
}